// Q_RUNLayer_Hybrid_14542759265071
// MI455X (gfx1250) — compile-verified
//
#include <hip/hip_runtime.h>

#define TOK   16
#define PROJD 1024
#define NBK   16
#define BSZ   64
#define RANKD 64
#define HID   4096

typedef float v8f __attribute__((ext_vector_type(8)));
typedef float v2f __attribute__((ext_vector_type(2)));

__device__ __forceinline__ v8f wmma4(v2f a, v2f b, v8f c) {
  // D(16x16,f32) = A(16x4,f32) * B(4x16,f32) + C
  return __builtin_amdgcn_wmma_f32_16x16x4_f32(false, a, false, b, (short)0, c, false, false);
}

__device__ __forceinline__ v8f splat8(float v) {
  v8f r = {v, v, v, v, v, v, v, v};
  return r;
}

__global__ __launch_bounds__(256) void qrun_fused(
    const float* __restrict__ x,      // [N,1024]
    const float* __restrict__ R,      // [16,64,64]
    const float* __restrict__ Lw,     // [16,64,64]
    const float* __restrict__ bias,   // [1024]
    const float* __restrict__ gamma,  // [1024]
    const float* __restrict__ beta,   // [1024]
    const float* __restrict__ Wd,     // [1024,64]
    const float* __restrict__ Wu,     // [64,1024]
    const float* __restrict__ thetas, // [3,1024]
    const float* __restrict__ phis,   // [3,1024]
    const float* __restrict__ Wp,     // [4,6]
    const float* __restrict__ bpv,    // [4]
    float* __restrict__ out)          // [N,4096]
{
  __shared__ float s_buf[TOK][PROJD];   // 64 KB: h_perm -> h2 (in place) -> h3 (in place)
  __shared__ float s_t[TOK][RANKD];     // 4 KB: low-rank bottleneck

  const int tid   = threadIdx.x;
  const int wave  = tid >> 5;           // 0..7 (wave32)
  const int lane  = tid & 31;
  const int lhalf = lane >> 4;          // 0: lanes 0-15, 1: lanes 16-31
  const int l16   = lane & 15;
  const long tile0 = (long)blockIdx.x * TOK;

  // hint: pull this tile's x into cache early (global_prefetch_b8)
  __builtin_prefetch(x + tile0 * PROJD + (long)tid * 32, 0, 3);

  // zero the rank-accumulation tile
  for (int i = tid; i < TOK * RANKD; i += 256) ((float*)s_t)[i] = 0.0f;

  // ---------------- Stage A: h1 = x * blockdiag(R), scattered permuted into s_buf ----
  // wave owns blocks b = 2*wave, 2*wave+1 ; PERM^-1(b*64+i) = i*16+b
  for (int bb = 0; bb < 2; ++bb) {
    const int b = wave * 2 + bb;
    const float* xblk = x + tile0 * PROJD + b * BSZ;
    const float* Rblk = R + b * BSZ * BSZ;
    v8f a0 = splat8(0.f), a1 = splat8(0.f), a2 = splat8(0.f), a3 = splat8(0.f);
    for (int k = 0; k < 16; ++k) {
      const int krow = k * 4 + lhalf * 2;           // K index of this lane-half
      v2f afr = *(const v2f*)(xblk + (long)l16 * PROJD + krow);
      const float* bp0 = Rblk + krow * BSZ + l16;
      v2f b0 = {bp0[0],  bp0[BSZ]};
      v2f b1 = {bp0[16], bp0[16 + BSZ]};
      v2f b2 = {bp0[32], bp0[32 + BSZ]};
      v2f b3 = {bp0[48], bp0[48 + BSZ]};
      a0 = wmma4(afr, b0, a0);
      a1 = wmma4(afr, b1, a1);
      a2 = wmma4(afr, b2, a2);
      a3 = wmma4(afr, b3, a3);
    }
#pragma unroll
    for (int r = 0; r < 8; ++r) {
      const int M = r + lhalf * 8;
      s_buf[M][(0 * 16 + l16) * 16 + b] = a0[r];
      s_buf[M][(1 * 16 + l16) * 16 + b] = a1[r];
      s_buf[M][(2 * 16 + l16) * 16 + b] = a2[r];
      s_buf[M][(3 * 16 + l16) * 16 + b] = a3[r];
    }
  }
  __syncthreads();

  // ---------------- Stage B: h2 = h_perm * blockdiag(L) + bias, in place --------------
  for (int bb = 0; bb < 2; ++bb) {
    const int b = wave * 2 + bb;
    const float* Lblk = Lw + b * BSZ * BSZ;
    v8f a0 = splat8(bias[b * BSZ +  0 + l16]);
    v8f a1 = splat8(bias[b * BSZ + 16 + l16]);
    v8f a2 = splat8(bias[b * BSZ + 32 + l16]);
    v8f a3 = splat8(bias[b * BSZ + 48 + l16]);
    for (int k = 0; k < 16; ++k) {
      const int krow = k * 4 + lhalf * 2;
      v2f afr = *(const v2f*)(&s_buf[l16][b * BSZ + krow]);
      const float* bp0 = Lblk + krow * BSZ + l16;
      v2f b0 = {bp0[0],  bp0[BSZ]};
      v2f b1 = {bp0[16], bp0[16 + BSZ]};
      v2f b2 = {bp0[32], bp0[32 + BSZ]};
      v2f b3 = {bp0[48], bp0[48 + BSZ]};
      a0 = wmma4(afr, b0, a0);
      a1 = wmma4(afr, b1, a1);
      a2 = wmma4(afr, b2, a2);
      a3 = wmma4(afr, b3, a3);
    }
#pragma unroll
    for (int r = 0; r < 8; ++r) {
      const int M = r + lhalf * 8;
      s_buf[M][b * BSZ +  0 + l16] = a0[r];
      s_buf[M][b * BSZ + 16 + l16] = a1[r];
      s_buf[M][b * BSZ + 32 + l16] = a2[r];
      s_buf[M][b * BSZ + 48 + l16] = a3[r];
    }
  }
  __syncthreads();

  // ---------------- LayerNorm (2 tokens per wave, wave32 shuffle reduce) ---------------
  for (int tt = 0; tt < 2; ++tt) {
    const int n = wave * 2 + tt;
    float s = 0.f, ss = 0.f;
    for (int i = 0; i < 32; ++i) {
      const float v = s_buf[n][lane + 32 * i];
      s += v; ss += v * v;
    }
#pragma unroll
    for (int off = 16; off; off >>= 1) {
      s  += __shfl_xor(s,  off, 32);
      ss += __shfl_xor(ss, off, 32);
    }
    const float mu = s * (1.0f / PROJD);
    const float rs = rsqrtf(ss * (1.0f / PROJD) - mu * mu + 1e-5f);
    for (int i = 0; i < 32; ++i) {
      const int c = lane + 32 * i;
      const float v = s_buf[n][c];
      s_buf[n][c] = (v - mu) * rs * gamma[c] + beta[c];
    }
  }
  __syncthreads();

  // ---------------- Low-rank down: s_t = h2 @ Wd (K split across waves) ----------------
  {
    const int kbase = wave * 128;
    v8f a0 = splat8(0.f), a1 = splat8(0.f), a2 = splat8(0.f), a3 = splat8(0.f);
    for (int k = 0; k < 32; ++k) {
      const int krow = kbase + k * 4 + lhalf * 2;
      v2f afr = *(const v2f*)(&s_buf[l16][krow]);
      const float* bp0 = Wd + krow * RANKD + l16;
      v2f b0 = {bp0[0],  bp0[RANKD]};
      v2f b1 = {bp0[16], bp0[16 + RANKD]};
      v2f b2 = {bp0[32], bp0[32 + RANKD]};
      v2f b3 = {bp0[48], bp0[48 + RANKD]};
      a0 = wmma4(afr, b0, a0);
      a1 = wmma4(afr, b1, a1);
      a2 = wmma4(afr, b2, a2);
      a3 = wmma4(afr, b3, a3);
    }
#pragma unroll
    for (int r = 0; r < 8; ++r) {
      const int M = r + lhalf * 8;
      atomicAdd(&s_t[M][ 0 + l16], a0[r]);   // ds_add_f32
      atomicAdd(&s_t[M][16 + l16], a1[r]);
      atomicAdd(&s_t[M][32 + l16], a2[r]);
      atomicAdd(&s_t[M][48 + l16], a3[r]);
    }
  }
  __syncthreads();

  // ---------------- Low-rank up + residual: h3 = h2 + s_t @ Wu (in place) -------------
  for (int jj = 0; jj < 8; ++jj) {
    const int jt   = wave * 8 + jj;        // 0..63 column tiles of 16
    const int colb = jt * 16 + l16;
    v8f acc;
#pragma unroll
    for (int r = 0; r < 8; ++r) acc[r] = s_buf[r + lhalf * 8][colb];   // residual as C
    for (int k = 0; k < 16; ++k) {
      const int krow = k * 4 + lhalf * 2;
      v2f afr = *(const v2f*)(&s_t[l16][krow]);
      const float* bp0 = Wu + krow * PROJD + colb;
      v2f bf = {bp0[0], bp0[PROJD]};
      acc = wmma4(afr, bf, acc);
    }
#pragma unroll
    for (int r = 0; r < 8; ++r) s_buf[r + lhalf * 8][colb] = acc[r];
  }
  __syncthreads();

  // ---------------- Tail: rotations, sin/cos, 6->4 linear, coalesced float4 stores ----
  float w0[6], w1[6], w2[6], w3[6];
#pragma unroll
  for (int j = 0; j < 6; ++j) {
    w0[j] = Wp[j]; w1[j] = Wp[6 + j]; w2[j] = Wp[12 + j]; w3[j] = Wp[18 + j];
  }
  const float c0 = bpv[0], c1 = bpv[1], c2 = bpv[2], c3 = bpv[3];

  for (int i = 0; i < 64; ++i) {
    const int idx = tid + 256 * i;        // 0..16383 over [16 tokens][1024 cols]
    const int n = idx >> 10;
    const int p = idx & 1023;
    const float h = s_buf[n][p];
    float o0 = c0, o1 = c1, o2 = c2, o3 = c3;
#pragma unroll
    for (int r = 0; r < 3; ++r) {
      const float rot = fmaf(h, thetas[r * PROJD + p], phis[r * PROJD + p]);
      const float sn = __sinf(rot);
      const float cs = __cosf(rot);
      o0 = fmaf(sn, w0[2 * r], fmaf(cs, w0[2 * r + 1], o0));
      o1 = fmaf(sn, w1[2 * r], fmaf(cs, w1[2 * r + 1], o1));
      o2 = fmaf(sn, w2[2 * r], fmaf(cs, w2[2 * r + 1], o2));
      o3 = fmaf(sn, w3[2 * r], fmaf(cs, w3[2 * r + 1], o3));
    }
    float4 o = make_float4(o0, o1, o2, o3);
    *(float4*)(out + (tile0 + n) * HID + p * 4) = o;   // out[n][4p+u]
  }
}

extern "C" void kernel_launch(void* const* d_in, const int* in_sizes, int n_in,
                              void* d_out, int out_size, void* d_ws, size_t ws_size,
                              hipStream_t stream) {
  const float* x      = (const float*)d_in[0];
  const float* R      = (const float*)d_in[1];
  const float* Lw     = (const float*)d_in[2];
  const float* bias   = (const float*)d_in[3];
  const float* gamma  = (const float*)d_in[4];
  const float* beta   = (const float*)d_in[5];
  const float* Wd     = (const float*)d_in[6];
  const float* Wu     = (const float*)d_in[7];
  const float* thetas = (const float*)d_in[8];
  const float* phis   = (const float*)d_in[9];
  const float* Wp     = (const float*)d_in[10];
  const float* bp     = (const float*)d_in[11];
  float* out = (float*)d_out;

  const int ntok = in_sizes[0] / PROJD;    // 8192
  dim3 grid(ntok / TOK), block(256);
  qrun_fused<<<grid, block, 0, stream>>>(x, R, Lw, bias, gamma, beta,
                                         Wd, Wu, thetas, phis, Wp, bp, out);
}